// LIFNodeTriton_25580825215390
// MI455X (gfx1250) — compile-verified
//
#include <hip/hip_runtime.h>
#include <stdint.h>

// LIF scan: x (N=32, T=16, C=128, H=32, W=32) f32 -> spikes same shape.
//   v = 0.5*v + x_t ; spike = (v >= 1.0) ; v -= spike
// Memory-bound (528 MB @ 23.3 TB/s ~ 23 us). Nonlinear recurrence -> no WMMA.
// Data path: CDNA5 async global->LDS staging (ASYNCcnt), in-order completion,
// software-pipelined with s_wait_asynccnt; non-temporal b128 stores.

typedef float v4f __attribute__((ext_vector_type(4)));

#define T_STEPS 16
#define CHW     131072            // 128*32*32 floats per (n,t) slice
#define CHW4    (CHW / 4)         // 32768 float4 per (n,t) slice
#define BLOCK   256

__global__ __launch_bounds__(BLOCK) void lif_scan_kernel(
    const float* __restrict__ x,
    const float* __restrict__ v_init,
    float* __restrict__ out)
{
    // 64 KB staging tile: [t][tid] -> one float4 per lane per timestep
    __shared__ v4f smem[T_STEPS * BLOCK];

    const int      tid = threadIdx.x;
    const unsigned pos = blockIdx.x * BLOCK + tid;   // float4 site in (N,C,H,W)
    const unsigned n   = pos >> 15;                  // pos / CHW4
    const unsigned r4  = pos & (CHW4 - 1);           // float4 index within CHW
    const unsigned r   = r4 << 2;                    // float index within CHW

    // Per-lane byte offset of element (n, t=0, r) inside x:
    //   bytes = (n*16*CHW + r) * 4 = (n << 23) + (r << 2)   (max ~2^28 < 2^31: ok as I32 voffset)
    const uint32_t voff_base = (n << 23) + (r << 2);
    // LDS byte address of this lane's t=0 slot (low 32 bits of flat shared addr = LDS offset)
    const uint32_t lds_base  = (uint32_t)(uintptr_t)(&smem[tid]);

    // Issue all 16 async global->LDS b128 loads up front (deep MLP, tracked by ASYNCcnt,
    // completing in order). t-stride in bytes: CHW*4 = 524288; LDS stride: BLOCK*16 = 4096.
    #pragma unroll
    for (int t = 0; t < T_STEPS; ++t) {
        uint32_t ldsa = lds_base + (uint32_t)t * (BLOCK * 16);
        uint32_t moff = voff_base + (uint32_t)t * (CHW * 4);
        asm volatile("global_load_async_to_lds_b128 %0, %1, %2"
                     :
                     : "v"(ldsa), "v"(moff), "s"(x)
                     : "memory");
    }

    // Membrane state (v_init is (N,C,H,W): flat float index = pos*4)
    v4f v = *(const v4f*)(v_init + (size_t)pos * 4);

    // Two-stage pipeline: async loads finish in order, so ASYNCcnt<=8 means t=0..7 landed.
    #pragma unroll
    for (int h = 0; h < 2; ++h) {
        if (h == 0) {
            asm volatile("s_wait_asynccnt 8" ::: "memory");
        } else {
            asm volatile("s_wait_asynccnt 0" ::: "memory");
        }
        #pragma unroll
        for (int j = 0; j < 8; ++j) {
            const int t = h * 8 + j;
            v4f xi = smem[t * BLOCK + tid];          // ds_load_b128 (own lane's slot)
            v = v * 0.5f + xi;                       // leaky integrate (exact: *0.5)
            v4f sp;
            #pragma unroll
            for (int k = 0; k < 4; ++k)
                sp[k] = (v[k] >= 1.0f) ? 1.0f : 0.0f; // fire
            v = v - sp;                               // soft reset (V_TH = 1)
            v4f* dst = (v4f*)(out + (size_t)(n * T_STEPS + t) * CHW + r);
            __builtin_nontemporal_store(sp, dst);     // streaming store, b128 NT
        }
    }
}

extern "C" void kernel_launch(void* const* d_in, const int* in_sizes, int n_in,
                              void* d_out, int out_size, void* d_ws, size_t ws_size,
                              hipStream_t stream) {
    const float* x      = (const float*)d_in[0];   // (32,16,128,32,32) f32
    const float* v_init = (const float*)d_in[1];   // (32,128,32,32)    f32
    float*       out    = (float*)d_out;           // (32,16,128,32,32) f32

    const int sites     = in_sizes[1];             // N*C*H*W = 4,194,304
    const int positions = sites / 4;               // float4 sites = 1,048,576
    const int blocks    = positions / BLOCK;       // 4096 (exact)

    lif_scan_kernel<<<blocks, BLOCK, 0, stream>>>(x, v_init, out);
}